// GAT_11278584119485
// MI455X (gfx1250) — compile-verified
//
#include <hip/hip_runtime.h>
#include <math.h>

// ---------------------------------------------------------------------------
// GAT forward on gfx1250 (CDNA5).
//  - Dense GEMMs: V_WMMA_F32_16X16X4_F32, register-blocked over column tiles.
//  - Edge aggregation: CSR-by-dst, one wave per node, coalesced gathers,
//    zero float atomics. Epilogues (bias+relu / bias+log_softmax) fused.
// ---------------------------------------------------------------------------

typedef float v2f __attribute__((ext_vector_type(2)));
typedef float v8f __attribute__((ext_vector_type(8)));

#define NFEAT   256
#define NHID    16
#define HEADS   8
#define NCLASS  40
#define F1      (HEADS * NHID)   // 128
#define NEG_SLOPE 0.2f
#define SCAN_B  256

__device__ __forceinline__ void edge_ends(int e, int E, const int* __restrict__ src,
                                          const int* __restrict__ dst, int& s, int& d) {
    if (e < E) { s = src[e]; d = dst[e]; }
    else       { s = e - E;  d = e - E;  }          // self loops appended
}

__device__ __forceinline__ float lrelu(float x) {
    return x > 0.f ? x : NEG_SLOPE * x;
}

// ======================= CSR build ==========================================
__global__ void gat_init(float* __restrict__ m1, float* __restrict__ s1,
                         float* __restrict__ m2, float* __restrict__ s2,
                         int* __restrict__ deg, int N) {
    int tid = blockIdx.x * blockDim.x + threadIdx.x;
    if (tid < N)         { m2[tid] = -INFINITY; s2[tid] = 0.f; deg[tid] = 0; }
    if (tid < N * HEADS) { m1[tid] = -INFINITY; s1[tid] = 0.f; }
}

__global__ void gat_deg(const int* __restrict__ src, const int* __restrict__ dst,
                        int E, int Etot, int* __restrict__ deg) {
    int e = blockIdx.x * blockDim.x + threadIdx.x;
    if (e >= Etot) return;
    int s, d; edge_ends(e, E, src, dst, s, d);
    atomicAdd(&deg[d], 1);
}

// per-block inclusive scan of deg
__global__ void gat_scan_blocks(const int* __restrict__ deg, int* __restrict__ incl,
                                int* __restrict__ bsums, int N) {
    __shared__ int sh[SCAN_B];
    int i = blockIdx.x * SCAN_B + threadIdx.x;
    int v = (i < N) ? deg[i] : 0;
    sh[threadIdx.x] = v;
    __syncthreads();
    for (int off = 1; off < SCAN_B; off <<= 1) {
        int t = (threadIdx.x >= off) ? sh[threadIdx.x - off] : 0;
        __syncthreads();
        sh[threadIdx.x] += t;
        __syncthreads();
    }
    if (i < N) incl[i] = sh[threadIdx.x];
    if (threadIdx.x == SCAN_B - 1) bsums[blockIdx.x] = sh[SCAN_B - 1];
}

// exclusive scan of block sums (single block; nb <= 1024)
__global__ void gat_scan_sums(const int* __restrict__ bsums, int* __restrict__ boff, int nb) {
    __shared__ int sh[1024];
    int t = threadIdx.x;
    int v = (t < nb) ? bsums[t] : 0;
    sh[t] = v;
    __syncthreads();
    for (int off = 1; off < 1024; off <<= 1) {
        int u = (t >= off) ? sh[t - off] : 0;
        __syncthreads();
        sh[t] += u;
        __syncthreads();
    }
    if (t < nb) boff[t] = sh[t] - v;   // exclusive
}

__global__ void gat_rowptr(const int* __restrict__ incl, const int* __restrict__ deg,
                           const int* __restrict__ boff, int* __restrict__ rowptr,
                           int* __restrict__ cursor, int N) {
    int i = blockIdx.x * blockDim.x + threadIdx.x;
    if (i >= N) return;
    int ex = incl[i] - deg[i] + boff[i >> 8];   // SCAN_B == 256
    rowptr[i] = ex;
    cursor[i] = ex;
    if (i == N - 1) rowptr[N] = ex + deg[i];
}

__global__ void gat_fill_adj(const int* __restrict__ src, const int* __restrict__ dst,
                             int E, int Etot, int* __restrict__ cursor,
                             int* __restrict__ adj) {
    int e = blockIdx.x * blockDim.x + threadIdx.x;
    if (e >= Etot) return;
    int s, d; edge_ends(e, E, src, dst, s, d);
    int pos = atomicAdd(&cursor[d], 1);
    adj[pos] = s;
}

// ======================= GEMMs (WMMA f32 16x16x4) ===========================
// h1[N,128] = x[N,256] @ W1[256,128]; 4 column tiles per wave.
__global__ void gat_gemm1(const float* __restrict__ X, const float* __restrict__ W,
                          float* __restrict__ H) {
    const int lane = threadIdx.x;          // 0..31
    const int hs   = lane >> 4;
    const int l    = lane & 15;
    const int row0 = blockIdx.x * 16;
    const int cb   = blockIdx.y * 64;      // 4 tiles of 16 cols

    v8f a0 = {}, a1 = {}, a2 = {}, a3 = {};
    const float* xrow = X + (size_t)(row0 + l) * NFEAT;
    #pragma unroll 2
    for (int k0 = 0; k0 < NFEAT; k0 += 4) {
        v2f a;
        a.x = xrow[k0 + 2 * hs];
        a.y = xrow[k0 + 2 * hs + 1];
        const float* w0 = W + (size_t)(k0 + 2 * hs)     * F1 + cb + l;
        const float* w1 = W + (size_t)(k0 + 2 * hs + 1) * F1 + cb + l;
        v2f b0, b1, b2, b3;
        b0.x = w0[0];  b0.y = w1[0];
        b1.x = w0[16]; b1.y = w1[16];
        b2.x = w0[32]; b2.y = w1[32];
        b3.x = w0[48]; b3.y = w1[48];
        a0 = __builtin_amdgcn_wmma_f32_16x16x4_f32(false, a, false, b0, (short)0, a0, false, false);
        a1 = __builtin_amdgcn_wmma_f32_16x16x4_f32(false, a, false, b1, (short)0, a1, false, false);
        a2 = __builtin_amdgcn_wmma_f32_16x16x4_f32(false, a, false, b2, (short)0, a2, false, false);
        a3 = __builtin_amdgcn_wmma_f32_16x16x4_f32(false, a, false, b3, (short)0, a3, false, false);
    }
    #pragma unroll
    for (int j = 0; j < 8; ++j) {
        float* o = H + (size_t)(row0 + j + hs * 8) * F1 + cb + l;
        o[0]  = a0[j];
        o[16] = a1[j];
        o[32] = a2[j];
        o[48] = a3[j];
    }
}

// h2[N,40] = out1[N,128] @ W2[128,40]; 3 column tiles per wave, last masked.
__global__ void gat_gemm2(const float* __restrict__ X, const float* __restrict__ W,
                          float* __restrict__ H) {
    const int lane = threadIdx.x;
    const int hs   = lane >> 4;
    const int l    = lane & 15;
    const int row0 = blockIdx.x * 16;
    const bool t2ok = (32 + l) < NCLASS;   // l < 8

    v8f a0 = {}, a1 = {}, a2 = {};
    const float* xrow = X + (size_t)(row0 + l) * F1;
    #pragma unroll 2
    for (int k0 = 0; k0 < F1; k0 += 4) {
        v2f a;
        a.x = xrow[k0 + 2 * hs];
        a.y = xrow[k0 + 2 * hs + 1];
        const float* w0 = W + (size_t)(k0 + 2 * hs)     * NCLASS + l;
        const float* w1 = W + (size_t)(k0 + 2 * hs + 1) * NCLASS + l;
        v2f b0, b1, b2;
        b0.x = w0[0];  b0.y = w1[0];
        b1.x = w0[16]; b1.y = w1[16];
        b2.x = t2ok ? w0[32] : 0.f;
        b2.y = t2ok ? w1[32] : 0.f;
        a0 = __builtin_amdgcn_wmma_f32_16x16x4_f32(false, a, false, b0, (short)0, a0, false, false);
        a1 = __builtin_amdgcn_wmma_f32_16x16x4_f32(false, a, false, b1, (short)0, a1, false, false);
        a2 = __builtin_amdgcn_wmma_f32_16x16x4_f32(false, a, false, b2, (short)0, a2, false, false);
    }
    #pragma unroll
    for (int j = 0; j < 8; ++j) {
        float* o = H + (size_t)(row0 + j + hs * 8) * NCLASS + l;
        o[0]  = a0[j];
        o[16] = a1[j];
        if (t2ok) o[32] = a2[j];
    }
}

// ======================= attention logits ===================================
__global__ void gat_alpha1(const float* __restrict__ H, const float* __restrict__ att_s,
                           const float* __restrict__ att_d,
                           float* __restrict__ as1, float* __restrict__ ad1, int N) {
    int tid = blockIdx.x * blockDim.x + threadIdx.x;
    if (tid >= N * HEADS) return;
    int n = tid >> 3, h = tid & 7;
    const float* hp = H + (size_t)n * F1 + h * NHID;
    float a = 0.f, b = 0.f;
    #pragma unroll
    for (int d = 0; d < NHID; ++d) {
        a += hp[d] * att_s[h * NHID + d];
        b += hp[d] * att_d[h * NHID + d];
    }
    as1[tid] = a; ad1[tid] = b;
}

__global__ void gat_alpha2(const float* __restrict__ H, const float* __restrict__ att_s,
                           const float* __restrict__ att_d,
                           float* __restrict__ as2, float* __restrict__ ad2, int N) {
    int n = blockIdx.x * blockDim.x + threadIdx.x;
    if (n >= N) return;
    const float* hp = H + (size_t)n * NCLASS;
    float a = 0.f, b = 0.f;
    #pragma unroll
    for (int c = 0; c < NCLASS; ++c) { a += hp[c] * att_s[c]; b += hp[c] * att_d[c]; }
    as2[n] = a; ad2[n] = b;
}

// ======================= softmax max/sum (small atomics) ====================
__global__ void gat_emax1(const float* __restrict__ as1, const float* __restrict__ ad1,
                          const int* __restrict__ src, const int* __restrict__ dst,
                          int E, int Etot, float* __restrict__ m1) {
    int tid = blockIdx.x * blockDim.x + threadIdx.x;
    if (tid >= Etot * HEADS) return;
    int e = tid >> 3, h = tid & 7;
    int s, d; edge_ends(e, E, src, dst, s, d);
    atomicMax(&m1[d * HEADS + h], lrelu(as1[s * HEADS + h] + ad1[d * HEADS + h]));
}

__global__ void gat_esum1(const float* __restrict__ as1, const float* __restrict__ ad1,
                          const float* __restrict__ m1,
                          const int* __restrict__ src, const int* __restrict__ dst,
                          int E, int Etot, float* __restrict__ s1) {
    int tid = blockIdx.x * blockDim.x + threadIdx.x;
    if (tid >= Etot * HEADS) return;
    int e = tid >> 3, h = tid & 7;
    int s, d; edge_ends(e, E, src, dst, s, d);
    float ev = lrelu(as1[s * HEADS + h] + ad1[d * HEADS + h]);
    atomicAdd(&s1[d * HEADS + h], __expf(ev - m1[d * HEADS + h]));
}

__global__ void gat_emax2(const float* __restrict__ as2, const float* __restrict__ ad2,
                          const int* __restrict__ src, const int* __restrict__ dst,
                          int E, int Etot, float* __restrict__ m2) {
    int e = blockIdx.x * blockDim.x + threadIdx.x;
    if (e >= Etot) return;
    int s, d; edge_ends(e, E, src, dst, s, d);
    atomicMax(&m2[d], lrelu(as2[s] + ad2[d]));
}

__global__ void gat_esum2(const float* __restrict__ as2, const float* __restrict__ ad2,
                          const float* __restrict__ m2,
                          const int* __restrict__ src, const int* __restrict__ dst,
                          int E, int Etot, float* __restrict__ s2) {
    int e = blockIdx.x * blockDim.x + threadIdx.x;
    if (e >= Etot) return;
    int s, d; edge_ends(e, E, src, dst, s, d);
    float ev = lrelu(as2[s] + ad2[d]);
    atomicAdd(&s2[d], __expf(ev - m2[d]));
}

// ======================= CSR aggregation (no atomics) =======================
// One wave per dst node; lanes cover 128 features as float4 chunks.
// Fused epilogue: out1 = relu(agg + b1).
__global__ void gat_agg1_csr(const float* __restrict__ H, const float* __restrict__ as1,
                             const float* __restrict__ ad1, const float* __restrict__ m1,
                             const float* __restrict__ s1, const int* __restrict__ rowptr,
                             const int* __restrict__ adj, const float* __restrict__ b1,
                             float* __restrict__ out1, int N) {
    int d = blockIdx.x * blockDim.y + threadIdx.y;
    if (d >= N) return;
    int lane = threadIdx.x;            // feature chunk 0..31
    int h    = lane >> 2;              // head
    float adh = ad1[d * HEADS + h];
    float mdh = m1[d * HEADS + h];
    float sdh = s1[d * HEADS + h] + 1e-16f;
    float4 acc = {0.f, 0.f, 0.f, 0.f};
    int k1 = rowptr[d + 1];
    for (int k = rowptr[d]; k < k1; ++k) {
        int s = adj[k];                                 // uniform across wave
        float w = __expf(lrelu(as1[s * HEADS + h] + adh) - mdh) / sdh;
        float4 hv = *(const float4*)(H + (size_t)s * F1 + lane * 4);
        acc.x += w * hv.x; acc.y += w * hv.y;
        acc.z += w * hv.z; acc.w += w * hv.w;
    }
    int c = lane * 4;
    float* o = out1 + (size_t)d * F1 + c;
    o[0] = fmaxf(acc.x + b1[c + 0], 0.f);
    o[1] = fmaxf(acc.y + b1[c + 1], 0.f);
    o[2] = fmaxf(acc.z + b1[c + 2], 0.f);
    o[3] = fmaxf(acc.w + b1[c + 3], 0.f);
}

// Layer 2: one wave per dst node; lanes cover 40 classes (lane, lane+32).
// Fused epilogue: bias + log_softmax via wave shuffle reduction -> d_out.
__global__ void gat_agg2_csr(const float* __restrict__ H, const float* __restrict__ as2,
                             const float* __restrict__ ad2, const float* __restrict__ m2,
                             const float* __restrict__ s2, const int* __restrict__ rowptr,
                             const int* __restrict__ adj, const float* __restrict__ b2,
                             float* __restrict__ out, int N) {
    int d = blockIdx.x * blockDim.y + threadIdx.y;
    if (d >= N) return;
    int lane = threadIdx.x;
    bool hi  = lane < (NCLASS - 32);   // lane < 8
    float add = ad2[d], md = m2[d], sd = s2[d] + 1e-16f;
    float acc0 = 0.f, acc1 = 0.f;
    int k1 = rowptr[d + 1];
    for (int k = rowptr[d]; k < k1; ++k) {
        int s = adj[k];
        float w = __expf(lrelu(as2[s] + add) - md) / sd;
        const float* hp = H + (size_t)s * NCLASS;
        acc0 += w * hp[lane];
        if (hi) acc1 += w * hp[32 + lane];
    }
    float v0 = acc0 + b2[lane];
    float v1 = hi ? (acc1 + b2[32 + lane]) : -INFINITY;
    float mx = fmaxf(v0, v1);
    #pragma unroll
    for (int off = 16; off >= 1; off >>= 1) mx = fmaxf(mx, __shfl_xor(mx, off));
    float sm = __expf(v0 - mx) + (hi ? __expf(v1 - mx) : 0.f);
    #pragma unroll
    for (int off = 16; off >= 1; off >>= 1) sm += __shfl_xor(sm, off);
    float lse = mx + __logf(sm);
    out[(size_t)d * NCLASS + lane] = v0 - lse;
    if (hi) out[(size_t)d * NCLASS + 32 + lane] = v1 - lse;
}

// ---------------------------------------------------------------------------
extern "C" void kernel_launch(void* const* d_in, const int* in_sizes, int n_in,
                              void* d_out, int out_size, void* d_ws, size_t ws_size,
                              hipStream_t stream) {
    const float* x        = (const float*)d_in[0];
    const int*   eidx     = (const int*)  d_in[1];
    const float* W1       = (const float*)d_in[2];
    const float* att_src1 = (const float*)d_in[3];
    const float* att_dst1 = (const float*)d_in[4];
    const float* b1       = (const float*)d_in[5];
    const float* W2       = (const float*)d_in[6];
    const float* att_src2 = (const float*)d_in[7];
    const float* att_dst2 = (const float*)d_in[8];
    const float* b2       = (const float*)d_in[9];
    float*       out      = (float*)d_out;

    const int N    = in_sizes[0] / NFEAT;   // 50000 (divisible by 16)
    const int E    = in_sizes[1] / 2;       // 800000
    const int Etot = E + N;
    const int* src = eidx;
    const int* dst = eidx + E;

    // ---- workspace carve-out ----
    float* fws = (float*)d_ws;
    float* h1   = fws;                         // N*128
    float* out1 = h1   + (size_t)N * F1;       // N*128
    float* as1  = out1 + (size_t)N * F1;       // N*8
    float* ad1  = as1  + (size_t)N * HEADS;
    float* m1   = ad1  + (size_t)N * HEADS;
    float* s1   = m1   + (size_t)N * HEADS;
    float* as2  = s1   + (size_t)N * HEADS;    // N each
    float* ad2  = as2  + N;
    float* m2   = ad2  + N;
    float* s2   = m2   + N;
    float* h2   = h1;                          // N*40, aliases dead h1
    int*   iws    = (int*)(s2 + N);
    int*   deg    = iws;                       // N
    int*   cursor = deg    + N;                // N
    int*   incl   = cursor + N;                // N
    int*   boff   = incl   + N;                // 1024
    int*   rowptr = boff   + 1024;             // N+1
    int*   adj    = rowptr + (N + 1);          // Etot

    const int B = 256;
    auto blocks = [](long long n, int b) { return (unsigned)((n + b - 1) / b); };
    const int nScan = (int)blocks(N, SCAN_B);
    const dim3 nodeBlk(32, 8);                 // 1 wave per node, 8 nodes/block
    const unsigned nodeGrd = blocks(N, 8);

    // ---- CSR build + inits ----
    gat_init<<<blocks((long long)N * HEADS, B), B, 0, stream>>>(m1, s1, m2, s2, deg, N);
    gat_deg<<<blocks(Etot, B), B, 0, stream>>>(src, dst, E, Etot, deg);
    gat_scan_blocks<<<nScan, SCAN_B, 0, stream>>>(deg, incl, boff /*tmp bsums*/, N);
    gat_scan_sums<<<1, 1024, 0, stream>>>(boff, boff, nScan);
    gat_rowptr<<<blocks(N, B), B, 0, stream>>>(incl, deg, boff, rowptr, cursor, N);
    gat_fill_adj<<<blocks(Etot, B), B, 0, stream>>>(src, dst, E, Etot, cursor, adj);

    // ---- layer 1 ----
    gat_gemm1<<<dim3(N / 16, F1 / 64), 32, 0, stream>>>(x, W1, h1);
    gat_alpha1<<<blocks((long long)N * HEADS, B), B, 0, stream>>>(h1, att_src1, att_dst1, as1, ad1, N);
    gat_emax1<<<blocks((long long)Etot * HEADS, B), B, 0, stream>>>(as1, ad1, src, dst, E, Etot, m1);
    gat_esum1<<<blocks((long long)Etot * HEADS, B), B, 0, stream>>>(as1, ad1, m1, src, dst, E, Etot, s1);
    gat_agg1_csr<<<nodeGrd, nodeBlk, 0, stream>>>(h1, as1, ad1, m1, s1, rowptr, adj, b1, out1, N);

    // ---- layer 2 ----
    gat_gemm2<<<N / 16, 32, 0, stream>>>(out1, W2, h2);
    gat_alpha2<<<blocks(N, B), B, 0, stream>>>(h2, att_src2, att_dst2, as2, ad2, N);
    gat_emax2<<<blocks(Etot, B), B, 0, stream>>>(as2, ad2, src, dst, E, Etot, m2);
    gat_esum2<<<blocks(Etot, B), B, 0, stream>>>(as2, ad2, m2, src, dst, E, Etot, s2);
    gat_agg2_csr<<<nodeGrd, nodeBlk, 0, stream>>>(h2, as2, ad2, m2, s2, rowptr, adj, b2, out, N);
}